// MultiHeadLiftLayer_67319317397857
// MI455X (gfx1250) — compile-verified
//
#include <hip/hip_runtime.h>

typedef float v2f __attribute__((ext_vector_type(2)));
typedef float v4f __attribute__((ext_vector_type(4)));
typedef float v8f __attribute__((ext_vector_type(8)));

#define IN_CH0 128
#define HEADS  8
#define IN_CH1 64
// output row = HEADS + IN_CH1 = 72 floats = 18 float4 (288 B, 16B-aligned rows)

// ---------------------------------------------------------------------------
// Kernel 1: node projection with fp32 WMMA.
//   P[node, 0: 8] = x0[node,:] @ att[:128, :]   (source half of W)
//   P[node, 8:16] = x0[node,:] @ att[128:, :]   (target half of W)
// One wave handles a 16-row tile; K=128 consumed as 32x V_WMMA_F32_16X16X4_F32.
// A 16x4 f32 layout: lanes 0-15 hold (M=lane, K=0/1), lanes 16-31 (M=lane-16, K=2/3).
// B  4x16 f32 layout mirrors it: lanes 0-15 hold (N=lane, K=0/1), lanes 16-31 K=2/3.
// ---------------------------------------------------------------------------
__global__ __launch_bounds__(256) void node_proj_wmma(
        const float* __restrict__ x0, const float* __restrict__ att,
        float* __restrict__ P, int num_nodes, int num_tiles) {
    const int lane = threadIdx.x & 31;
    const int tile = blockIdx.x * (blockDim.x >> 5) + (threadIdx.x >> 5);
    if (tile >= num_tiles) return;                 // wave-uniform exit: EXEC stays full
    const int base = tile * 16;
    const int half = lane >> 4;                    // 0 -> K=0,1 ; 1 -> K=2,3
    const int lrow = lane & 15;

    int arowi = base + lrow;
    if (arowi >= num_nodes) arowi = num_nodes - 1; // clamp, keep EXEC all-1s
    const float* arow = x0 + (size_t)arowi * IN_CH0 + 2 * half;

    // Column n of virtual Wcat[128][16]: Wcat[k][n] = (n<8) ? att[k*8+n]
    //                                                       : att[(128+k)*8 + n-8]
    const int n = lrow;
    const float* bcol = (n < HEADS) ? (att + n)
                                    : (att + (size_t)IN_CH0 * HEADS + (n - HEADS));

    v8f acc = {};
    #pragma unroll
    for (int k = 0; k < IN_CH0; k += 4) {
        v2f a = *(const v2f*)(arow + k);           // (K = k+2*half, k+1+2*half)
        const int kk = k + 2 * half;
        v2f b;
        b.x = bcol[(size_t)kk * HEADS];            // att is tiny -> L0/L2 resident
        b.y = bcol[(size_t)(kk + 1) * HEADS];
        acc = __builtin_amdgcn_wmma_f32_16x16x4_f32(
                /*neg_a=*/false, a, /*neg_b=*/false, b,
                /*c_mod=*/(short)0, acc, /*reuse_a=*/false, /*reuse_b=*/false);
    }

    // C/D layout: lane<16 -> (M = base+v,   N = lane)
    //             lane>=16 -> (M = base+8+v, N = lane-16)
    const int N = lrow;
    const int mbase = base + half * 8;
    #pragma unroll
    for (int v = 0; v < 8; ++v) {
        const int m = mbase + v;
        if (m < num_nodes) P[(size_t)m * 16 + N] = acc[v];
    }
}

// ---------------------------------------------------------------------------
// Kernel 2: streaming edge assembly. One thread per output float4.
//   j in [0,2):  out[e, 4j..4j+3] = relu(P[src[e], 4j..] + P[tgt[e], 8+4j..])
//   j in [2,18): out[e, 4j..4j+3] = x1[e, 4j-8 .. ]
// Output + x1 traffic is non-temporal so the L2 keeps P / x0 / att resident.
// ---------------------------------------------------------------------------
__global__ __launch_bounds__(256) void edge_assemble(
        const int* __restrict__ src, const int* __restrict__ tgt,
        const float* __restrict__ P, const float* __restrict__ x1,
        float* __restrict__ out, int num_edges) {
    const unsigned idx = blockIdx.x * blockDim.x + threadIdx.x;
    const unsigned total = (unsigned)num_edges * 18u;
    if (idx >= total) return;
    const unsigned e = idx / 18u;                  // magic-multiply division
    const unsigned j = idx - e * 18u;
    v4f* out4 = (v4f*)out;
    if (j < 2u) {
        const int s = src[e];
        const int t = tgt[e];
        const v4f ps = *(const v4f*)(P + (size_t)s * 16 + j * 4);
        const v4f pt = *(const v4f*)(P + (size_t)t * 16 + 8 + j * 4);
        v4f r;
        r.x = fmaxf(ps.x + pt.x, 0.0f);
        r.y = fmaxf(ps.y + pt.y, 0.0f);
        r.z = fmaxf(ps.z + pt.z, 0.0f);
        r.w = fmaxf(ps.w + pt.w, 0.0f);
        __builtin_nontemporal_store(r, out4 + idx);
    } else {
        const v4f v = __builtin_nontemporal_load(
                (const v4f*)x1 + (size_t)e * (IN_CH1 / 4) + (j - 2u));
        __builtin_nontemporal_store(v, out4 + idx);
    }
}

extern "C" void kernel_launch(void* const* d_in, const int* in_sizes, int n_in,
                              void* d_out, int out_size, void* d_ws, size_t ws_size,
                              hipStream_t stream) {
    const float* x0  = (const float*)d_in[0];   // [num_nodes, 128] fp32
    const int*   adj = (const int*)d_in[1];     // [2, num_edges] int
    const float* x1  = (const float*)d_in[2];   // [num_edges, 64] fp32
    const float* att = (const float*)d_in[3];   // [256, 8] fp32

    const int num_nodes = in_sizes[0] / IN_CH0;
    const int num_edges = in_sizes[1] / 2;
    float* P   = (float*)d_ws;                  // [num_nodes, 16] fp32 (3.2 MB)
    float* out = (float*)d_out;                 // [num_edges, 72] fp32

    // Kernel 1: one wave per 16-row tile, 8 waves (256 threads) per block.
    const int num_tiles = (num_nodes + 15) / 16;
    const int blocks1 = (num_tiles + 7) / 8;
    node_proj_wmma<<<dim3(blocks1), dim3(256), 0, stream>>>(x0, att, P,
                                                            num_nodes, num_tiles);

    // Kernel 2: one thread per output float4 (18 per edge).
    const unsigned total = (unsigned)num_edges * 18u;
    const unsigned blocks2 = (total + 255u) / 256u;
    edge_assemble<<<dim3(blocks2), dim3(256), 0, stream>>>(
            adj, adj + num_edges, P, x1, out, num_edges);
}